// OnlinePGHI_66073776882009
// MI455X (gfx1250) — compile-verified
//
#include <hip/hip_runtime.h>
#include <math.h>

// ---------------------------------------------------------------------------
// OnlinePGHI on MI455X (gfx1250), single-wave32 kernel.
//
// Reference collapses (M2=1): fgradw/mag_buffer unused; 1-D segmented
// trapezoidal integration of dphase_t, seeded at each segment's first argmax.
// phase[i] = prefix[i] - prefix[seed(i)], prefix = cumsum(0.5*(t[i-1]+t[i])).
// Row-wise 16-wide prefix sums done with V_WMMA_F32_16X16X4_F32 (x4 K-chunks),
// fp32 end to end.
// ---------------------------------------------------------------------------

typedef __attribute__((ext_vector_type(2))) float v2f;
typedef __attribute__((ext_vector_type(8))) float v8f;

#define PGHI_NB     511
#define PGHI_ABSTOL 1e-10f

// Compile-time double constants, cast to float for the f32 pipeline.
constexpr double D_PI    = 3.14159265358979323846264338327950288;
constexpr double D_LN100 = 4.60517018598809136803598290936872841; // -ln(0.01)
constexpr double D_GAMMA = 2.0 * D_PI * (1048576.0 / (8.0 * D_LN100)); // N_FFT^2 = 1048576
constexpr double D_FMUL  = D_GAMMA / (2.0 * 256.0 * 1024.0);           // HOP=256, N_FFT=1024
constexpr double D_C1    = 1.0 / (4.0 * D_FMUL);
constexpr double D_C2    = 2.0 * D_PI * 256.0 / 1024.0;                // = pi/2 per-k step

__global__ __launch_bounds__(32) void OnlinePGHI_66073776882009_kernel(
    const float* __restrict__ x,   // 513 floats
    float* __restrict__ out)       // 511 floats
{
    __shared__ float s_xl[513];    // log(x)
    __shared__ float s_t[512];     // dphase_t (padded)
    __shared__ float s_d[512];     // trapezoid increments
    __shared__ float s_m[512];     // mag (padded with barrier at 511)
    __shared__ float s_p[512];     // row prefixes -> global prefix
    __shared__ float s_rowsum[32];
    __shared__ int   s_seed[512];

    const int lane = threadIdx.x;          // 0..31, one wave32
    const int m    = lane & 15;            // row (A) / column (B,D) index
    const int hi   = lane >> 4;            // lane half selects K / M sub-block

    // ---- Stage 1: logs ----------------------------------------------------
    for (int i = lane; i < 513; i += 32) s_xl[i] = logf(x[i]);
    __syncthreads();

    // ---- Stage 2: t, mag --------------------------------------------------
    const float C1 = (float)D_C1;
    const float C2 = (float)D_C2;
    for (int i = lane; i < 512; i += 32) {
        float t = 0.0f, mg = 0.0f;
        if (i < PGHI_NB) {
            t  = C1 * (s_xl[i + 2] - s_xl[i]) + C2 * (float)(i + 1);
            mg = s_xl[i + 1];
        }
        s_t[i] = t;
        s_m[i] = mg;                       // i==511 -> 0 (acts as barrier sentinel)
    }
    __syncthreads();

    // ---- Stage 3: trapezoid increments d ---------------------------------
    for (int i = lane; i < 512; i += 32)
        s_d[i] = (i == 0) ? 0.0f : 0.5f * (s_t[i - 1] + s_t[i]);
    __syncthreads();

    // ---- Stage 4: 16-wide row prefixes via WMMA f32 16x16x4 ---------------
    // Block b holds rows 16b..16b+15; element i = 16*row + col.
    // D = A x U, U upper-triangular ones => D[m][n] = sum_{k<=n} A[m][k].
    for (int b = 0; b < 2; ++b) {
        v8f acc = {};
        for (int kb = 0; kb < 4; ++kb) {
            const int kA = 4 * kb + 2 * hi;           // this lane's 2 K-columns
            v2f a, u;
            a.x = s_d[256 * b + 16 * m + kA];
            a.y = s_d[256 * b + 16 * m + kA + 1];
            u.x = (kA     <= m) ? 1.0f : 0.0f;        // U[kA][n],  n = lane&15
            u.y = (kA + 1 <= m) ? 1.0f : 0.0f;        // U[kA+1][n]
            acc = __builtin_amdgcn_wmma_f32_16x16x4_f32(
                false, a, false, u, (short)0, acc, false, false);
        }
        // D layout: VGPR j, lanes0-15 -> M=j, lanes16-31 -> M=j+8, N=lane&15.
        #pragma unroll
        for (int j = 0; j < 8; ++j) {
            const int row = j + 8 * hi;
            s_p[256 * b + 16 * row + m] = acc[j];
        }
    }
    __syncthreads();

    // ---- Stage 5: cross-row carry (32 row sums, exclusive scan) -----------
    s_rowsum[lane] = s_p[16 * lane + 15];
    __syncthreads();
    float off = 0.0f;
    for (int r = 0; r < lane; ++r) off += s_rowsum[r];
    #pragma unroll
    for (int n = 0; n < 16; ++n) s_p[16 * lane + n] += off;   // own row only
    __syncthreads();

    // ---- Stage 6: segment seeds (first argmax per active run) -------------
    if (lane == 0) {
        float curmax = -INFINITY;
        int   curarg = -1, segstart = 0;
        for (int i = 0; i < 512; ++i) {
            const float mg = s_m[i];
            if (!(mg > PGHI_ABSTOL)) {                 // barrier / sentinel
                for (int j = segstart; j < i; ++j) s_seed[j] = curarg;
                s_seed[i] = -1;
                segstart = i + 1;
                curmax = -INFINITY;
                curarg = -1;
            } else if (mg > curmax) {                  // strict > == first argmax
                curmax = mg;
                curarg = i;
            }
        }
    }
    __syncthreads();

    // ---- Stage 7: phase = prefix[i] - prefix[seed]; 0 on barriers ---------
    for (int i = lane; i < PGHI_NB; i += 32) {
        const int s = s_seed[i];
        out[i] = (s < 0) ? 0.0f : (s_p[i] - s_p[s]);
    }
}

extern "C" void kernel_launch(void* const* d_in, const int* in_sizes, int n_in,
                              void* d_out, int out_size, void* d_ws, size_t ws_size,
                              hipStream_t stream) {
    (void)in_sizes; (void)n_in; (void)d_ws; (void)ws_size; (void)out_size;
    const float* x = (const float*)d_in[0];     // (1,1,513) f32
    // d_in[1] (mag_buffer) provably unused: M2==1 kills N/S neighbors.
    float* out = (float*)d_out;                 // (1,1,511) f32
    OnlinePGHI_66073776882009_kernel<<<dim3(1), dim3(32), 0, stream>>>(x, out);
}